// DiffusionInitializer_34454227648701
// MI455X (gfx1250) — compile-verified
//
#include <hip/hip_runtime.h>

// CDNA5 / gfx1250. Wave32. V_WMMA_F32_16X16X4_F32 for the [BS,1024]x[1024,16]
// (N padded 3->16) projection. B fragments pre-swizzled into WMMA layout in LDS
// so the inner loop is: clause of global_load_b64 (A) + ds_load_2addr_b64 (B) + wmma.
// Diffusion scan collapsed analytically: x = c*noise + (1-c)*target, c = prod(i/steps).

typedef __attribute__((ext_vector_type(2))) float v2f;
typedef __attribute__((ext_vector_type(8))) float v8f;

#define D_DIM            1024
#define K_CHUNKS         (D_DIM / 4)                         // 256 WMMAs per tile
#define ROWS_PER_WAVE    16
#define WAVES_PER_BLOCK  8
#define ROWS_PER_BLOCK   (ROWS_PER_WAVE * WAVES_PER_BLOCK)   // 128
#define BLOCK_THREADS    (WAVES_PER_BLOCK * 32)              // 256

__global__ __launch_bounds__(BLOCK_THREADS)
void diffusion_init_wmma_kernel(const float* __restrict__ latent,  // [rows, D]
                                const float* __restrict__ W,       // [D, 3]
                                const float* __restrict__ bvec,    // [3]
                                const float* __restrict__ noise,   // [rows, 3]
                                const int*   __restrict__ steps_p, // [1]
                                float*       __restrict__ out,     // [rows, 3]
                                int total_rows)
{
    // B fragments in WMMA 4x16 f32 register layout, one float2 per (chunk, lane).
    // 256 chunks * 32 lanes * 8B = 64 KB LDS (320 KB/WGP -> ~5 blocks resident).
    __shared__ v2f w_frag[K_CHUNKS * 32];

    // Cooperative fill: entry (chunk, lane) = {Wpad[k][col], Wpad[k+1][col]},
    // k = 4*chunk + 2*(lane>>4), col = lane&15, Wpad[., col>=3] = 0.
    for (int idx = threadIdx.x; idx < K_CHUNKS * 32; idx += BLOCK_THREADS) {
        const int chunk = idx >> 5;
        const int ln    = idx & 31;
        const int fhi   = ln >> 4;
        const int fcol  = ln & 15;
        const int k     = 4 * chunk + 2 * fhi;
        v2f f;
        f.x = (fcol < 3) ? W[k * 3 + fcol]       : 0.0f;
        f.y = (fcol < 3) ? W[(k + 1) * 3 + fcol] : 0.0f;
        w_frag[idx] = f;
    }
    __syncthreads();

    const int lane = threadIdx.x & 31;
    const int wave = threadIdx.x >> 5;
    const int hi   = lane >> 4;    // A: K-pair select; C/D: M+8 select
    const int col  = lane & 15;    // A: row-in-tile; B/C/D: N column

    const int rowBase = blockIdx.x * ROWS_PER_BLOCK + wave * ROWS_PER_WAVE;
    if (rowBase >= total_rows) return;  // wave-uniform exit (EXEC all-ones for WMMA)

    // A-fragment source row (clamped for a possible ragged final tile).
    int aRow = rowBase + col;
    if (aRow > total_rows - 1) aRow = total_rows - 1;
    const float* aPtr = latent + (size_t)aRow * D_DIM + 2 * hi;
    const v2f*   bPtr = w_frag + lane;

    v8f acc = {};  // C/D 16x16 f32 accumulator (8 VGPRs)

    // Inner loop: one 16x16x4 f32 WMMA per 4 elements of K.
    #pragma unroll 8
    for (int c = 0; c < K_CHUNKS; ++c) {
        v2f a = *(const v2f*)(aPtr + 4 * c);   // global_load_b64, streaming
        v2f b = bPtr[(size_t)c * 32];          // ds_load_b64, unconditional
        acc = __builtin_amdgcn_wmma_f32_16x16x4_f32(
            /*neg_a=*/false, a, /*neg_b=*/false, b,
            /*c_mod=*/(short)0, acc, /*reuse_a=*/false, /*reuse_b=*/false);
    }

    // Collapse the affine scan: coefficients sum to 1, so x = cn*noise + (1-cn)*target.
    // One divide total; loop body is cvt + two muls (no per-iteration division).
    const int steps = *steps_p;
    const float inv_steps = 1.0f / (float)steps;
    float cn = 1.0f;
    for (int i = steps; i >= 1; --i)
        cn *= (float)i * inv_steps;
    const float ct = 1.0f - cn;

    // Epilogue: C/D layout — VGPR v holds M = v + 8*hi, N = col. Only N<3 is real.
    if (col < 3) {
        const float bb = bvec[col];
        #pragma unroll
        for (int v = 0; v < 8; ++v) {
            const int row = rowBase + v + 8 * hi;
            if (row < total_rows) {
                const float target = acc[v] + bb;
                const size_t idx = (size_t)row * 3 + col;
                out[idx] = cn * noise[idx] + ct * target;
            }
        }
    }
}

extern "C" void kernel_launch(void* const* d_in, const int* in_sizes, int n_in,
                              void* d_out, int out_size, void* d_ws, size_t ws_size,
                              hipStream_t stream)
{
    const float* latent  = (const float*)d_in[0];  // [B,S,D] f32
    const float* W       = (const float*)d_in[1];  // [D,3]   f32
    const float* bvec    = (const float*)d_in[2];  // [3]     f32
    const float* noise   = (const float*)d_in[3];  // [B,S,3] f32
    const int*   steps_p = (const int*)d_in[4];    // [1]     int

    float* out = (float*)d_out;                    // [B,S,3] f32

    const int total_rows = in_sizes[0] / D_DIM;    // B*S
    const int grid = (total_rows + ROWS_PER_BLOCK - 1) / ROWS_PER_BLOCK;

    diffusion_init_wmma_kernel<<<grid, BLOCK_THREADS, 0, stream>>>(
        latent, W, bvec, noise, steps_p, out, total_rows);
}